// HybridFFNN_81896436400511
// MI455X (gfx1250) — compile-verified
//
#include <hip/hip_runtime.h>
#include <math.h>

typedef float v2f __attribute__((ext_vector_type(2)));
typedef float v8f __attribute__((ext_vector_type(8)));

#if defined(__has_builtin)
#if __has_builtin(__builtin_amdgcn_tanhf)
#define FAST_TANH(x) __builtin_amdgcn_tanhf(x)
#else
#define FAST_TANH(x) tanhf(x)
#endif
#else
#define FAST_TANH(x) tanhf(x)
#endif

// D = A(16x4 f32) * B(4x16 f32) + C(16x16 f32)
__device__ __forceinline__ v8f wmma_f32(v2f a, v2f b, v8f c) {
  return __builtin_amdgcn_wmma_f32_16x16x4_f32(
      /*neg_a=*/false, a, /*neg_b=*/false, b,
      /*c_mod=*/(short)0, c, /*reuse_a=*/false, /*reuse_b=*/false);
}

// One hidden layer: H_out = act(H_in @ W^T + b), staged through per-wave LDS.
// LDS tile layout: hl[token*10 + feature], stride 10 -> conflict-free & 8B aligned.
__device__ __forceinline__ void hidden_layer(float* hl, const v2f bw[2], float bias,
                                             int nout, bool do_tanh,
                                             int n, int h2, int m0) {
  v8f c = {};
#pragma unroll
  for (int g = 0; g < 2; ++g) {
    int k = 4 * g + h2;
    v2f a;
    a.x = hl[n * 10 + k];
    a.y = hl[n * 10 + k + 1];
    c = wmma_f32(a, bw[g], c);
  }
  __builtin_amdgcn_wave_barrier();   // keep loads (old H) before stores (new H)
  if (n < nout) {
#pragma unroll
    for (int r = 0; r < 8; ++r) {
      float v = c[r] + bias;
      hl[(r + m0) * 10 + n] = do_tanh ? FAST_TANH(v) : v;
    }
  }
  __builtin_amdgcn_wave_barrier();
}

__global__ __launch_bounds__(256) void hybrid_ffnn_wmma_kernel(
    const float* __restrict__ x, const float* __restrict__ S,
    const float* __restrict__ W1, const float* __restrict__ b1,
    const float* __restrict__ W2, const float* __restrict__ b2,
    const float* __restrict__ W3, const float* __restrict__ b3,
    const float* __restrict__ W4, const float* __restrict__ b4,
    const float* __restrict__ W5, const float* __restrict__ b5,
    float* __restrict__ out, int n_tiles) {
  __shared__ float lds[8 * 160];              // 8 waves * (16 tokens * 10 floats)
  const int lane = threadIdx.x & 31;
  const int wv   = threadIdx.x >> 5;
  const int n    = lane & 15;                 // A row (token) / B,D column
  const int h2   = (lane >> 4) << 1;          // K sub-offset within a K=4 group
  const int m0   = (lane >> 4) << 3;          // D row offset (0 or 8)
  float* hl = &lds[wv * 160];

  // ---- Build B fragments once per wave (weights in WMMA B layout, zero-padded) ----
  v2f bw1[7];
#pragma unroll
  for (int g = 0; g < 7; ++g)
#pragma unroll
    for (int j = 0; j < 2; ++j) {
      int k = 4 * g + h2 + j;
      bw1[g][j] = (n < 8 && k < 27) ? W1[n * 27 + k] : 0.f;
    }
  v2f bw2[2], bw3[2], bw4[2], bw5[2], bs0[2], bs1[2];
#pragma unroll
  for (int g = 0; g < 2; ++g)
#pragma unroll
    for (int j = 0; j < 2; ++j) {
      int k = 4 * g + h2 + j;                 // 0..7
      bw2[g][j] = (n < 8) ? W2[n * 8 + k] : 0.f;
      bw3[g][j] = (n < 8) ? W3[n * 8 + k] : 0.f;
      bw4[g][j] = (n < 7) ? W4[n * 8 + k] : 0.f;
      bw5[g][j] = (n < 7 && k < 7) ? W5[n * 7 + k] : 0.f;
      bs0[g][j] = (k < 7) ? S[n * 7 + k] : 0.f;               // cols 0..15 of S^T
      bs1[g][j] = (16 + n < 27 && k < 7) ? S[(16 + n) * 7 + k] : 0.f; // cols 16..26
    }
  const float bias1 = (n < 8) ? b1[n] : 0.f;
  const float bias2 = (n < 8) ? b2[n] : 0.f;
  const float bias3 = (n < 8) ? b3[n] : 0.f;
  const float bias4 = (n < 7) ? b4[n] : 0.f;
  const float bias5 = (n < 7) ? b5[n] : 0.f;

  const long stride = (long)gridDim.x * 8;
  for (long tile = (long)blockIdx.x * 8 + wv; tile < (long)n_tiles; tile += stride) {
    const float* xb = x + tile * (16 * 27);
    if (tile + stride < (long)n_tiles)
      __builtin_prefetch(x + (tile + stride) * (16 * 27), 0, 1);

    // ---- Layer 1: A straight from global (16 tokens x 27, K padded to 28) ----
    v8f acc = {};
#pragma unroll
    for (int g = 0; g < 7; ++g) {
      int k = 4 * g + h2;
      v2f a;
      a.x = xb[n * 27 + k];
      a.y = (k + 1 < 27) ? xb[n * 27 + k + 1] : 0.f;
      acc = wmma_f32(a, bw1[g], acc);
    }
    if (n < 8) {
#pragma unroll
      for (int r = 0; r < 8; ++r)
        hl[(r + m0) * 10 + n] = FAST_TANH(acc[r] + bias1);
    }
    __builtin_amdgcn_wave_barrier();

    // ---- Layers 2..5 through LDS (same-wave DS ops are in-order) ----
    hidden_layer(hl, bw2, bias2, 8, true,  n, h2, m0);
    hidden_layer(hl, bw3, bias3, 8, true,  n, h2, m0);
    hidden_layer(hl, bw4, bias4, 7, true,  n, h2, m0);
    hidden_layer(hl, bw5, bias5, 7, false, n, h2, m0);   // scores, no tanh

    // ---- Final projection: scores(16x7) @ S^T(7x27) -> two N tiles ----
    v2f af[2];
#pragma unroll
    for (int g = 0; g < 2; ++g) {
      int k = 4 * g + h2;
      af[g].x = (k < 7) ? hl[n * 10 + k] : 0.f;
      af[g].y = (k + 1 < 7) ? hl[n * 10 + k + 1] : 0.f;
    }
    __builtin_amdgcn_wave_barrier();
    v8f d0 = {}, d1 = {};
#pragma unroll
    for (int g = 0; g < 2; ++g) {
      d0 = wmma_f32(af[g], bs0[g], d0);
      d1 = wmma_f32(af[g], bs1[g], d1);
    }

    // ---- Store: D layout gives contiguous 64B runs per half-wave ----
    float* ob = out + tile * (16 * 27);
#pragma unroll
    for (int r = 0; r < 8; ++r)
      ob[(r + m0) * 27 + n] = d0[r];
    if (n < 11) {
#pragma unroll
      for (int r = 0; r < 8; ++r)
        ob[(r + m0) * 27 + 16 + n] = d1[r];
    }
  }
}

extern "C" void kernel_launch(void* const* d_in, const int* in_sizes, int n_in,
                              void* d_out, int out_size, void* d_ws, size_t ws_size,
                              hipStream_t stream) {
  const float* x  = (const float*)d_in[0];
  const float* S  = (const float*)d_in[1];
  const float* W1 = (const float*)d_in[2];
  const float* b1 = (const float*)d_in[3];
  const float* W2 = (const float*)d_in[4];
  const float* b2 = (const float*)d_in[5];
  const float* W3 = (const float*)d_in[6];
  const float* b3 = (const float*)d_in[7];
  const float* W4 = (const float*)d_in[8];
  const float* b4 = (const float*)d_in[9];
  const float* W5 = (const float*)d_in[10];
  const float* b5 = (const float*)d_in[11];
  float* out = (float*)d_out;

  long n_tokens = (long)in_sizes[0] / 27;     // B * SL = 1,048,576 (multiple of 16)
  int n_tiles = (int)(n_tokens / 16);         // 65,536 tiles of 16 tokens
  int blocks = 1024;                          // 8192 waves, ~8 tiles/wave
  if (blocks * 8 > n_tiles) blocks = (n_tiles + 7) / 8;
  if (blocks < 1) blocks = 1;

  hybrid_ffnn_wmma_kernel<<<blocks, 256, 0, stream>>>(
      x, S, W1, b1, W2, b2, W3, b3, W4, b4, W5, b5, out, n_tiles);
}